// Filter_29042568855669
// MI455X (gfx1250) — compile-verified
//
#include <hip/hip_runtime.h>
#include <cmath>

typedef float v2f __attribute__((ext_vector_type(2)));
typedef float v8f __attribute__((ext_vector_type(8)));
typedef int   v4i __attribute__((vector_size(16)));   // matches async builtin pointee

#define NMAT 4096
#define BM 128
#define BN 128
#define BK 32

#if defined(__has_builtin)
#  if __has_builtin(__builtin_amdgcn_global_load_async_to_lds_b128)
#    define USE_ASYNC_LDS 1
#  endif
#endif

typedef __attribute__((address_space(1))) v4i* gptr_b128;
typedef __attribute__((address_space(3))) v4i* lptr_b128;

__device__ __forceinline__ void async_cp16(const float* gsrc, float* ldst)
{
#ifdef USE_ASYNC_LDS
    __builtin_amdgcn_global_load_async_to_lds_b128(
        (gptr_b128)gsrc, (lptr_b128)ldst, 0, 0);
#else
    *(float4*)ldst = *(const float4*)gsrc;
#endif
}

__device__ __forceinline__ void async_wait_all()
{
#ifdef USE_ASYNC_LDS
#  if __has_builtin(__builtin_amdgcn_s_wait_asynccnt)
    __builtin_amdgcn_s_wait_asynccnt(0);
#  else
    asm volatile("s_wait_asynccnt 0x0" ::: "memory");
#  endif
#endif
}

// ---------------------------------------------------------------------------
// Init: T0 = I, T1 = L - I, out[f] = c0eff[f]*I + c1eff[f]*T1  (scale folded)
// ---------------------------------------------------------------------------
__global__ __launch_bounds__(256)
void cheb_init_kernel(const float* __restrict__ Lm, float* __restrict__ T0,
                      float* __restrict__ T1, float* __restrict__ out,
                      float4 c0, float4 c1)
{
    const size_t NN = (size_t)NMAT * NMAT;
    size_t i = ((size_t)blockIdx.x * blockDim.x + threadIdx.x) * 4;
    int r = (int)(i / NMAT);
    int c = (int)(i % NMAT);
    float4 lv = *(const float4*)(Lm + i);
    float d0 = (r == c    ) ? 1.0f : 0.0f;
    float d1 = (r == c + 1) ? 1.0f : 0.0f;
    float d2 = (r == c + 2) ? 1.0f : 0.0f;
    float d3 = (r == c + 3) ? 1.0f : 0.0f;
    float4 t1 = make_float4(lv.x - d0, lv.y - d1, lv.z - d2, lv.w - d3);
    *(float4*)(T0 + i) = make_float4(d0, d1, d2, d3);
    *(float4*)(T1 + i) = t1;
    *(float4*)(out + 0*NN + i) = make_float4(c0.x*d0 + c1.x*t1.x, c0.x*d1 + c1.x*t1.y,
                                             c0.x*d2 + c1.x*t1.z, c0.x*d3 + c1.x*t1.w);
    *(float4*)(out + 1*NN + i) = make_float4(c0.y*d0 + c1.y*t1.x, c0.y*d1 + c1.y*t1.y,
                                             c0.y*d2 + c1.y*t1.z, c0.y*d3 + c1.y*t1.w);
    *(float4*)(out + 2*NN + i) = make_float4(c0.z*d0 + c1.z*t1.x, c0.z*d1 + c1.z*t1.y,
                                             c0.z*d2 + c1.z*t1.z, c0.z*d3 + c1.z*t1.w);
    *(float4*)(out + 3*NN + i) = make_float4(c0.w*d0 + c1.w*t1.x, c0.w*d1 + c1.w*t1.y,
                                             c0.w*d2 + c1.w*t1.z, c0.w*d3 + c1.w*t1.w);
}

// ---------------------------------------------------------------------------
// One recurrence step, lower-triangular blocks only (everything is symmetric):
//   acc = L @ Tc            (fp32 WMMA GEMM, 128x128x32 tiles, async-LDS DMA)
//   t   = 2*(acc - Tc) - To (== (2L-2I)@Tc - To) ; Tn tile + out[f] += ck[f]*t
//   mirror tile (transpose via LDS) written for off-diagonal blocks
// ---------------------------------------------------------------------------
__global__ __launch_bounds__(256)
void cheb_step_kernel(const float* __restrict__ Lm, const float* __restrict__ Tc,
                      const float* __restrict__ To, float* __restrict__ Tn,
                      float* __restrict__ out, float4 ck)
{
    const int bx = blockIdx.x, by = blockIdx.y;
    if (by < bx) return;                       // symmetric: compute lower triangle

    __shared__ float smem[2 * BM * BK + 2 * BK * BN];   // 64 KB
    float* As = smem;                 // [2][BM*BK] raw L tile (double buffered)
    float* Bs = smem + 2 * BM * BK;   // [2][BK*BN] raw Tc tile (double buffered)

    const int tid  = threadIdx.x;
    const int lane = tid & 31;
    const int wave = tid >> 5;
    const int wm   = wave & 1;    // wave row  (2 waves over M)
    const int wn   = wave >> 1;   // wave col  (4 waves over N)
    const int hf   = lane >> 4;   // half-wave: K sub-pair select
    const int l16  = lane & 15;

    const int row0 = by * BM;
    const int col0 = bx * BN;

    const v8f vzero = {0.f,0.f,0.f,0.f,0.f,0.f,0.f,0.f};
    v8f acc[4][2];
    #pragma unroll
    for (int mt = 0; mt < 4; mt++)
        #pragma unroll
        for (int nt = 0; nt < 2; nt++) acc[mt][nt] = vzero;

    auto issueStage = [&](int buf, int kb) {
        #pragma unroll
        for (int rep = 0; rep < 4; rep++) {               // A: L[row0:+128, kb:+32]
            int idx = (tid + rep * 256) * 4;
            int ar  = idx / BK;
            int ak  = idx % BK;
            async_cp16(Lm + (size_t)(row0 + ar) * NMAT + kb + ak,
                       &As[buf * BM * BK + ar * BK + ak]);
        }
        #pragma unroll
        for (int rep = 0; rep < 4; rep++) {               // B: Tc[kb:+32, col0:+128]
            int idx = (tid + rep * 256) * 4;
            int br  = idx / BN;
            int bc  = idx % BN;
            async_cp16(Tc + (size_t)(kb + br) * NMAT + col0 + bc,
                       &Bs[buf * BK * BN + br * BN + bc]);
        }
    };

    issueStage(0, 0);
    async_wait_all();
    __syncthreads();

    for (int kb = 0; kb < NMAT; kb += BK) {
        const int buf = (kb / BK) & 1;
        if (kb + BK < NMAT) issueStage(buf ^ 1, kb + BK);

        #pragma unroll
        for (int ks = 0; ks < BK / 4; ks++) {
            const int kk = ks * 4;
            v2f afr[4];
            #pragma unroll
            for (int mt = 0; mt < 4; mt++) {
                int r = wm * 64 + mt * 16 + l16;
                afr[mt] = *(const v2f*)(&As[buf * BM * BK + r * BK + kk + 2 * hf]);
            }
            v2f bfr[2];
            #pragma unroll
            for (int nt = 0; nt < 2; nt++) {
                int c = wn * 32 + nt * 16 + l16;
                bfr[nt].x = Bs[buf * BK * BN + (kk + 2 * hf    ) * BN + c];
                bfr[nt].y = Bs[buf * BK * BN + (kk + 2 * hf + 1) * BN + c];
            }
            #pragma unroll
            for (int mt = 0; mt < 4; mt++)
                #pragma unroll
                for (int nt = 0; nt < 2; nt++)
                    acc[mt][nt] = __builtin_amdgcn_wmma_f32_16x16x4_f32(
                        false, afr[mt], false, bfr[nt],
                        (short)0, acc[mt][nt], false, false);
        }
        async_wait_all();
        __syncthreads();
    }

    // ---- Direct epilogue: t = 2*(acc - Tc) - To, kept in acc for mirror phase
    const size_t NN = (size_t)NMAT * NMAT;
    #pragma unroll
    for (int mt = 0; mt < 4; mt++) {
        #pragma unroll
        for (int nt = 0; nt < 2; nt++) {
            #pragma unroll
            for (int r = 0; r < 8; r++) {
                int gr = row0 + wm * 64 + mt * 16 + r + hf * 8;
                int gc = col0 + wn * 32 + nt * 16 + l16;
                size_t idx = (size_t)gr * NMAT + gc;
                float t = 2.0f * (acc[mt][nt][r] - Tc[idx]) - To[idx];
                acc[mt][nt][r] = t;
                Tn[idx] = t;
                out[idx]          += ck.x * t;
                out[NN + idx]     += ck.y * t;
                out[2 * NN + idx] += ck.z * t;
                out[3 * NN + idx] += ck.w * t;
            }
        }
    }

    if (by == bx) return;                 // diagonal tile: no mirror needed

    // ---- Mirror epilogue: transpose 32-column chunks through LDS (reuse smem)
    float* Tr = smem;                     // [32][132] padded transpose staging
    #pragma unroll 1
    for (int cc = 0; cc < 4; cc++) {
        if (wn == cc) {                   // this wave owns columns [cc*32, cc*32+32)
            #pragma unroll
            for (int mt = 0; mt < 4; mt++)
                #pragma unroll
                for (int nt = 0; nt < 2; nt++)
                    #pragma unroll
                    for (int r = 0; r < 8; r++) {
                        int gr_l = wm * 64 + mt * 16 + r + hf * 8;   // 0..127
                        int gc_l = nt * 16 + l16;                     // 0..31
                        Tr[gc_l * 132 + gr_l] = acc[mt][nt][r];
                    }
        }
        __syncthreads();
        #pragma unroll
        for (int rep = 0; rep < 4; rep++) {
            int q   = tid + rep * 256;    // 0..1023 float4s of the 32x128 chunk
            int cl  = q >> 5;             // 0..31  (mirror row within chunk)
            int cr4 = (q & 31) * 4;       // 0..124 (mirror col, float4 aligned)
            float4 v = *(const float4*)(&Tr[cl * 132 + cr4]);
            size_t midx = (size_t)(col0 + cc * 32 + cl) * NMAT + row0 + cr4;
            *(float4*)(Tn + midx) = v;
            #pragma unroll
            for (int f = 0; f < 4; f++) {
                float cf = (f == 0) ? ck.x : (f == 1) ? ck.y : (f == 2) ? ck.z : ck.w;
                float4 o = *(const float4*)(out + f * NN + midx);
                o.x += cf * v.x; o.y += cf * v.y; o.z += cf * v.z; o.w += cf * v.w;
                *(float4*)(out + f * NN + midx) = o;
            }
        }
        __syncthreads();
    }
}

// ---------------------------------------------------------------------------
// Host: Chebyshev coefficients (compile-time constants -> deterministic CPU calc)
// ---------------------------------------------------------------------------
static void compute_coeffs(float cEff[33][4])
{
    const int m = 32, Nc = 33;
    const double a1 = 1.0, a2 = 1.0;
    const double taus[4] = {0.5, 1.0, 2.0, 4.0};
    const double scale = 64.0;  // sqrt(4096)
    for (int o = 0; o <= m; o++) {
        for (int f = 0; f < 4; f++) {
            double s = 0.0;
            for (int j = 0; j < Nc; j++) {
                double x = cos(M_PI * (j + 0.5) / Nc);
                double g = exp(-taus[f] * (a1 * x + a2));
                s += cos(M_PI * o * (j + 0.5) / Nc) * g;
            }
            double c = (2.0 / Nc) * s;
            if (o == 0) c *= 0.5;
            cEff[o][f] = (float)(c * scale);
        }
    }
}

extern "C" void kernel_launch(void* const* d_in, const int* in_sizes, int n_in,
                              void* d_out, int out_size, void* d_ws, size_t ws_size,
                              hipStream_t stream)
{
    const float* Lm = (const float*)d_in[0];
    float* out = (float*)d_out;

    const size_t planeElems = (size_t)NMAT * NMAT;
    const size_t planeBytes = planeElems * sizeof(float);

    float* T[3];
    T[0] = (float*)d_ws;                                   // T_{k-2}
    T[1] = (float*)((char*)d_ws + planeBytes);             // T_{k-1}
    T[2] = (float*)((char*)d_ws + 2 * planeBytes);         // T_k

    float cEff[33][4];
    compute_coeffs(cEff);

    const int initBlocks = (int)(planeElems / (256 * 4));  // 16384
    cheb_init_kernel<<<initBlocks, 256, 0, stream>>>(
        Lm, T[0], T[1], out,
        make_float4(cEff[0][0], cEff[0][1], cEff[0][2], cEff[0][3]),
        make_float4(cEff[1][0], cEff[1][1], cEff[1][2], cEff[1][3]));

    dim3 grid(NMAT / BN, NMAT / BM);   // 32 x 32 (upper-tri blocks exit early)
    int io = 0, ic = 1, in = 2;
    for (int k = 2; k <= 32; k++) {
        float4 ck = make_float4(cEff[k][0], cEff[k][1], cEff[k][2], cEff[k][3]);
        cheb_step_kernel<<<grid, 256, 0, stream>>>(Lm, T[ic], T[io], T[in], out, ck);
        int t = io; io = ic; ic = in; in = t;  // rotate buffers
    }
}